// SADSimilarity_38706245272206
// MI455X (gfx1250) — compile-verified
//
#include <hip/hip_runtime.h>

#ifndef __has_builtin
#define __has_builtin(x) 0
#endif

#if __has_builtin(__builtin_amdgcn_global_load_async_to_lds_b32) && \
    __has_builtin(__builtin_amdgcn_s_wait_asynccnt)
#define USE_ASYNC_LDS 1
#else
#define USE_ASYNC_LDS 0
#endif

typedef float v4f __attribute__((ext_vector_type(4)));
typedef __attribute__((address_space(1))) int* gptr_i32;
typedef __attribute__((address_space(3))) int* lptr_i32;

constexpr int kB = 4, kN = 1024, kM = 1024, kD = 64;
constexpr int TN = 64, TM = 64;      // output tile per workgroup
constexpr int THREADS = 256;         // 8 waves (wave32)

// Stage one fp32 element global -> LDS through the gfx1250 async engine.
// Per-lane LDS address VGPR lets the async unit perform the transpose.
__device__ __forceinline__ void stage_b32(const float* g, float* l) {
#if USE_ASYNC_LDS
  __builtin_amdgcn_global_load_async_to_lds_b32(
      (gptr_i32)g, (lptr_i32)l, /*imm offset*/ 0, /*cpol*/ 0);
#else
  *l = *g;
#endif
}

__global__ __launch_bounds__(THREADS) void sad64_kernel(
    const float* __restrict__ lhs, const float* __restrict__ rhs,
    float* __restrict__ out) {
  // Tiles stored TRANSPOSED: [d][n] and [d][m], stride 64 words.
  // Compute-phase b128 reads are bank-conflict-free.
  __shared__ float sL[kD * TN];
  __shared__ float sR[kD * TM];

  const int b   = blockIdx.z;
  const int n0  = blockIdx.y * TN;
  const int m0  = blockIdx.x * TM;
  const int tid = threadIdx.x;

  // ---- async stage: 4096 elems per tile, 16 per thread per tile ----
  {
    const int sn = tid & 63;   // n (or m) across lanes -> consecutive LDS banks
    const int dq = tid >> 6;   // 0..3
    const float* gL = lhs + ((size_t)b * kN + (n0 + sn)) * kD;
    const float* gR = rhs + ((size_t)b * kM + (m0 + sn)) * kD;
#pragma unroll
    for (int i = 0; i < 16; ++i) {
      const int d = dq + i * 4;
      stage_b32(gL + d, &sL[d * TN + sn]);
      stage_b32(gR + d, &sR[d * TM + sn]);
    }
  }
#if USE_ASYNC_LDS
  __builtin_amdgcn_s_wait_asynccnt(0);   // s_wait_asynccnt 0 (per-wave)
#endif
  __syncthreads();

  // ---- compute: 4x4 micro-tile per thread ----
  const int tx = tid & 15;   // m direction: 16 threads x 4 = 64
  const int ty = tid >> 4;   // n direction: 16 threads x 4 = 64
  const int mi = tx * 4;
  const int ni = ty * 4;

  float acc[4][4];
#pragma unroll
  for (int j = 0; j < 4; ++j)
#pragma unroll
    for (int i = 0; i < 4; ++i) acc[j][i] = 0.0f;

#pragma unroll 4
  for (int d = 0; d < kD; ++d) {
    // one ds_load_b128 each: 4 n-values / 4 m-values at this depth
    const v4f a = *(const v4f*)&sL[d * TN + ni];
    const v4f c = *(const v4f*)&sR[d * TM + mi];
#pragma unroll
    for (int j = 0; j < 4; ++j) {
#pragma unroll
      for (int i = 0; i < 4; ++i) {
        // sign folded into accumulation: v_sub_f32 + v_sub_f32 with |src|
        // modifier -> still 2 VALU / element, no epilogue negation needed
        acc[j][i] -= __builtin_fabsf(a[j] - c[i]);
      }
    }
  }

  // ---- coalesced nontemporal stores (output is streamed, never re-read) ----
  float* o = out + (((size_t)b * kN + (n0 + ni)) * kM + (m0 + mi));
#pragma unroll
  for (int j = 0; j < 4; ++j) {
    v4f r = { acc[j][0], acc[j][1], acc[j][2], acc[j][3] };
    __builtin_nontemporal_store(r, (v4f*)(o + (size_t)j * kM));
  }
}

extern "C" void kernel_launch(void* const* d_in, const int* in_sizes, int n_in,
                              void* d_out, int out_size, void* d_ws, size_t ws_size,
                              hipStream_t stream) {
  (void)in_sizes; (void)n_in; (void)out_size; (void)d_ws; (void)ws_size;
  const float* lhs = (const float*)d_in[0];
  const float* rhs = (const float*)d_in[1];
  float* out = (float*)d_out;
  dim3 grid(kM / TM, kN / TN, kB);
  sad64_kernel<<<grid, dim3(THREADS), 0, stream>>>(lhs, rhs, out);
}